// att_GNN_38740605010123
// MI455X (gfx1250) — compile-verified
//
#include <hip/hip_runtime.h>
#include <math.h>

typedef float f32x2 __attribute__((ext_vector_type(2)));
typedef float f32x8 __attribute__((ext_vector_type(8)));

#define FDIM 64
#define NUM_GRAPHS 64
#define NEG_SLOPE 0.2f
#define KMAX 128
#define KPAD 132          // padded LDS row stride (floats): stride%64==4 -> conflict-free b64 reads
#define BLK_WAVES 8

// ---------------- utility ----------------

__global__ void fill_kernel(float* __restrict__ p, float v, int n) {
  int i = blockIdx.x * blockDim.x + threadIdx.x;
  if (i < n) p[i] = v;
}

__device__ __forceinline__ void atomic_max_f32(float* addr, float val) {
  // valid for IEEE floats when buffer initialized to -inf
  if (val >= 0.f) atomicMax((int*)addr, __float_as_int(val));
  else            atomicMin((unsigned int*)addr, __float_as_uint(val));
}

// ---------------- fused dual fp32 WMMA GEMM ----------------
// xl[M x 64] = A[M x K] @ Wl + bl ;  xr[M x 64] = A[M x K] @ Wr + br
// One wave per 16-row strip; 8 column tiles (4 for Wl, 4 for Wr) accumulated
// simultaneously via V_WMMA_F32_16X16X4_F32. Weights staged transposed in LDS.
// A layout (32-bit A 16x4): lanes 0-15 hold row M=lane, K=k+{0,1}; lanes 16-31 row M=lane-16, K=k+{2,3}.
// B layout mirrored; C layout: lanes 0-15 N=lane M=r, lanes 16-31 N=lane-16 M=8+r.
__global__ void gemm_dual_wmma_f32(const float* __restrict__ A,
                                   const float* __restrict__ Wl,
                                   const float* __restrict__ bl,
                                   const float* __restrict__ Wr,
                                   const float* __restrict__ br,
                                   float* __restrict__ xl,
                                   float* __restrict__ xr,
                                   int M, int K) {
  __shared__ float WT[2 * FDIM * KPAD];   // [0..63]=Wl^T rows, [64..127]=Wr^T rows

  // cooperative transpose-load of both weight matrices (coalesced global reads)
  for (int idx = threadIdx.x; idx < K * FDIM; idx += blockDim.x) {
    int k = idx / FDIM;
    int col = idx - k * FDIM;
    WT[col * KPAD + k]          = Wl[idx];
    WT[(FDIM + col) * KPAD + k] = Wr[idx];
  }
  __syncthreads();

  const int lane  = threadIdx.x & 31;
  const int wave  = threadIdx.x >> 5;
  const int tilesM = (M + 15) >> 4;
  const int tm = blockIdx.x * BLK_WAVES + wave;
  if (tm >= tilesM) return;                  // wave-uniform exit

  const int r16   = lane & 15;
  const int khalf = (lane >> 4) << 1;        // 0 or 2
  int row = tm * 16 + r16;
  row = row < M ? row : M - 1;               // clamp: no divergence, padding rows never stored
  const float* __restrict__ Arow = A + (size_t)row * K + khalf;

  f32x8 acc[8];
#pragma unroll
  for (int t = 0; t < 8; ++t) acc[t] = (f32x8){};

  for (int k = 0; k < K; k += 4) {
    f32x2 a;
    a.x = Arow[k];                           // global_load_b64 (contiguous pair)
    a.y = Arow[k + 1];
#pragma unroll
    for (int t = 0; t < 8; ++t) {
      const float* wt = &WT[(t * 16 + r16) * KPAD + k + khalf];
      f32x2 b;
      b.x = wt[0];                           // ds_load_b64, conflict-free
      b.y = wt[1];
      acc[t] = __builtin_amdgcn_wmma_f32_16x16x4_f32(
          false, a, false, b, (short)0, acc[t], false, false);
    }
  }

  const int mbase = tm * 16 + ((lane >> 4) << 3);
#pragma unroll
  for (int t = 0; t < 8; ++t) {
    float* __restrict__ out = (t < 4) ? xl : xr;
    const float* __restrict__ bv = (t < 4) ? bl : br;
    const int col = (t & 3) * 16 + r16;
    const float bc = bv[col];
#pragma unroll
    for (int r = 0; r < 8; ++r) {
      int m = mbase + r;
      if (m < M) out[(size_t)m * FDIM + col] = acc[t][r] + bc;
    }
  }
}

// ---------------- edge kernels (wave32 per edge, 2 features per lane) ----------------

__global__ void edge_logit_kernel(const float* __restrict__ xl,
                                  const float* __restrict__ xr,
                                  const float* __restrict__ euclid,
                                  const int* __restrict__ src,
                                  const int* __restrict__ dst,
                                  const float* __restrict__ We,
                                  const float* __restrict__ att,
                                  float* __restrict__ logits,
                                  float* __restrict__ segmax,
                                  int E) {
  int gid  = blockIdx.x * blockDim.x + threadIdx.x;
  int e    = gid >> 5;
  int lane = threadIdx.x & 31;
  if (e >= E) return;
  int s = src[e], d = dst[e];
  float eu = euclid[e];
  float acc = 0.f;
#pragma unroll
  for (int j = 0; j < 2; ++j) {
    int f = lane + 32 * j;
    float m = xl[(size_t)s * FDIM + f] + xr[(size_t)d * FDIM + f] + eu * We[f];
    m = (m > 0.f) ? m : NEG_SLOPE * m;       // leaky_relu
    acc += m * att[f];
  }
#pragma unroll
  for (int off = 16; off > 0; off >>= 1)
    acc += __shfl_down(acc, off, 32);
  if (lane == 0) {
    logits[e] = acc;
    atomic_max_f32(&segmax[d], acc);
  }
}

__global__ void edge_exp_kernel(float* __restrict__ logits,
                                const float* __restrict__ segmax,
                                const int* __restrict__ dst,
                                float* __restrict__ denom, int E) {
  int e = blockIdx.x * blockDim.x + threadIdx.x;
  if (e >= E) return;
  int d = dst[e];
  float v = __expf(logits[e] - segmax[d]);
  logits[e] = v;                             // buffer now holds exp(logit - max)
  atomicAdd(&denom[d], v);
}

__global__ void edge_agg_kernel(const float* __restrict__ ee,
                                const float* __restrict__ denom,
                                const float* __restrict__ xl,
                                const int* __restrict__ src,
                                const int* __restrict__ dst,
                                float* __restrict__ agg, int E) {
  int gid  = blockIdx.x * blockDim.x + threadIdx.x;
  int e    = gid >> 5;
  int lane = threadIdx.x & 31;
  if (e >= E) return;
  int s = src[e], d = dst[e];
  float alpha = ee[e] / denom[d];
#pragma unroll
  for (int j = 0; j < 2; ++j) {
    int f = lane + 32 * j;
    atomicAdd(&agg[(size_t)d * FDIM + f], alpha * xl[(size_t)s * FDIM + f]);
  }
}

__global__ void bias_relu_kernel(float* __restrict__ agg,
                                 const float* __restrict__ c, int total) {
  int i = blockIdx.x * blockDim.x + threadIdx.x;
  if (i < total) {
    float v = agg[i] + c[i & (FDIM - 1)];
    agg[i] = v > 0.f ? v : 0.f;
  }
}

// ---------------- pooling + head ----------------

__global__ void pool_kernel(const float* __restrict__ h,
                            const int* __restrict__ batch,
                            float* __restrict__ pooled,
                            float* __restrict__ counts, int N) {
  int gid  = blockIdx.x * blockDim.x + threadIdx.x;
  int n    = gid >> 5;
  int lane = threadIdx.x & 31;
  if (n >= N) return;
  int g = batch[n];
#pragma unroll
  for (int j = 0; j < 2; ++j) {
    int f = lane + 32 * j;
    atomicAdd(&pooled[(size_t)g * FDIM + f], h[(size_t)n * FDIM + f]);
  }
  if (lane == 0) atomicAdd(&counts[g], 1.f);
}

__global__ void head_kernel(const float* __restrict__ pooled,
                            const float* __restrict__ counts,
                            const float* __restrict__ Wlin,
                            const float* __restrict__ blin,
                            float* __restrict__ out) {
  int g = threadIdx.x;
  if (g >= NUM_GRAPHS) return;
  float cnt = fmaxf(counts[g], 1.f);
  float l0 = blin[0], l1 = blin[1];
  for (int f = 0; f < FDIM; ++f) {
    float p = pooled[(size_t)g * FDIM + f] / cnt;
    l0 += p * Wlin[f * 2 + 0];
    l1 += p * Wlin[f * 2 + 1];
  }
  float mx = fmaxf(l0, l1);
  float e0 = __expf(l0 - mx), e1 = __expf(l1 - mx);
  float s = e0 + e1;
  out[g * 2 + 0] = e0 / s;
  out[g * 2 + 1] = e1 / s;
}

// ---------------- launcher ----------------

extern "C" void kernel_launch(void* const* d_in, const int* in_sizes, int n_in,
                              void* d_out, int out_size, void* d_ws, size_t ws_size,
                              hipStream_t stream) {
  (void)n_in; (void)out_size; (void)ws_size;

  const float* x      = (const float*)d_in[0];   // [N, 128]
  const int*   ei     = (const int*)  d_in[1];   // [2, E] (int32 after jax x64-off downcast)
  const float* euclid = (const float*)d_in[2];   // [E, 1]
  const int*   batch  = (const int*)  d_in[3];   // [N]

  const int E  = in_sizes[2];
  const int N  = in_sizes[3];
  const int K1 = in_sizes[0] / N;                // 128
  const int* src = ei;
  const int* dst = ei + E;

  const float *Wl[3], *bl[3], *Wr[3], *br[3], *We[3], *att[3], *cb[3];
  for (int i = 0; i < 3; ++i) {
    int base = 4 + 7 * i;
    Wl[i]  = (const float*)d_in[base + 0];
    bl[i]  = (const float*)d_in[base + 1];
    Wr[i]  = (const float*)d_in[base + 2];
    br[i]  = (const float*)d_in[base + 3];
    We[i]  = (const float*)d_in[base + 4];
    att[i] = (const float*)d_in[base + 5];
    cb[i]  = (const float*)d_in[base + 6];
  }
  const float* Wlin = (const float*)d_in[25];
  const float* blin = (const float*)d_in[26];

  // ---- workspace carve-out ----
  char* ws = (char*)d_ws;
  size_t off = 0;
  auto carve = [&](size_t bytes) -> float* {
    float* p = (float*)(ws + off);
    off += (bytes + 255) & ~(size_t)255;
    return p;
  };
  float* buf0   = carve((size_t)N * FDIM * 4);
  float* buf1   = carve((size_t)N * FDIM * 4);
  float* xl     = carve((size_t)N * FDIM * 4);
  float* xr     = carve((size_t)N * FDIM * 4);
  float* logits = carve((size_t)E * 4);
  float* segmax = carve((size_t)N * 4);
  float* denom  = carve((size_t)N * 4);
  float* pooled = carve((size_t)(NUM_GRAPHS * FDIM + NUM_GRAPHS) * 4);
  float* counts = pooled + NUM_GRAPHS * FDIM;

  const int BT = 256;
  const int nodeBlk   = (N + BT - 1) / BT;
  const int featBlk   = (N * FDIM + BT - 1) / BT;
  const int edgeBlk   = (E + BT - 1) / BT;
  const int edgeWvBlk = (int)(((long long)E * 32 + BT - 1) / BT);
  const int tilesM    = (N + 15) / 16;
  const int gemmBlk   = (tilesM + BLK_WAVES - 1) / BLK_WAVES;

  const float* hin = x;
  float* aggbuf[3] = { buf0, buf1, buf0 };      // ping-pong h buffers
  int Ks[3] = { K1, FDIM, FDIM };

  for (int L = 0; L < 3; ++L) {
    float* agg = aggbuf[L];
    int K = Ks[L];

    fill_kernel<<<nodeBlk, BT, 0, stream>>>(segmax, -INFINITY, N);
    fill_kernel<<<nodeBlk, BT, 0, stream>>>(denom, 0.f, N);
    fill_kernel<<<featBlk, BT, 0, stream>>>(agg, 0.f, N * FDIM);

    gemm_dual_wmma_f32<<<gemmBlk, BT, 0, stream>>>(hin, Wl[L], bl[L], Wr[L], br[L],
                                                   xl, xr, N, K);

    edge_logit_kernel<<<edgeWvBlk, BT, 0, stream>>>(xl, xr, euclid, src, dst,
                                                    We[L], att[L], logits, segmax, E);
    edge_exp_kernel<<<edgeBlk, BT, 0, stream>>>(logits, segmax, dst, denom, E);
    edge_agg_kernel<<<edgeWvBlk, BT, 0, stream>>>(logits, denom, xl, src, dst, agg, E);
    bias_relu_kernel<<<featBlk, BT, 0, stream>>>(agg, cb[L], N * FDIM);

    hin = agg;
  }

  fill_kernel<<<(NUM_GRAPHS * FDIM + NUM_GRAPHS + BT - 1) / BT, BT, 0, stream>>>(
      pooled, 0.f, NUM_GRAPHS * FDIM + NUM_GRAPHS);
  pool_kernel<<<(int)(((long long)N * 32 + BT - 1) / BT), BT, 0, stream>>>(
      hin, batch, pooled, counts, N);
  head_kernel<<<1, 64, 0, stream>>>(pooled, counts, Wlin, blin, (float*)d_out);
}